// HashEmbedder1D_42262478193060
// MI455X (gfx1250) — compile-verified
//
#include <hip/hip_runtime.h>
#include <cmath>

// ---------------------------------------------------------------------------
// HashEmbedder1D: 2M samples x 16 levels, gather-2 + lerp, out [B, 32] f32.
// Memory-bound; level made wave-uniform (blockIdx.y) so all per-level
// constants and table pointers are scalar (SGPR), gathers use scalar-base +
// scale_offset addressing, and VALU work per sample is minimal. Exact f32
// divisions replaced by exact-reciprocal multiply + v_rcp_f32 (safe: the
// lerp is continuous across cell boundaries, so a 1-ulp floor flip changes
// the output by ~1e-10).
// ---------------------------------------------------------------------------

#define HE_N_LEVELS 16
#define HE_HASH_MASK ((1u << 19) - 1u)
#define HE_SAMPLES_PER_THREAD 4
#define HE_BLOCK 256

typedef float v2f __attribute__((ext_vector_type(2)));

struct HEParams {
    const v2f* tab[HE_N_LEVELS];    // per-level embedding tables [T_i][2] f32
    float      gs[HE_N_LEVELS];     // grid_size = 2^20 / res  (f32, matches ref)
    float      igs[HE_N_LEVELS];    // res * 2^-20 (exactly representable)
    unsigned   hashbits;            // bit i set -> level i uses hash mask
};

__global__ __launch_bounds__(HE_BLOCK)
void hashenc1d_kernel(const float* __restrict__ x,
                      v2f* __restrict__ out,
                      HEParams p,
                      unsigned nSamples)
{
    // Level is wave-uniform: every per-level quantity stays scalar.
    const unsigned lvl = blockIdx.y;
    const float    gs  = p.gs[lvl];
    const float    igs = p.igs[lvl];
    const v2f* __restrict__ t = p.tab[lvl];
    const unsigned m = ((p.hashbits >> lvl) & 1u) ? HE_HASH_MASK : 0xFFFFFFFFu;
    v2f* __restrict__ orow = out + lvl;          // + b*16 per sample

    const unsigned base = blockIdx.x * (HE_BLOCK * HE_SAMPLES_PER_THREAD)
                        + threadIdx.x;

#pragma unroll
    for (int k = 0; k < HE_SAMPLES_PER_THREAD; ++k) {
        unsigned b = base + (unsigned)k * HE_BLOCK;   // lane-consecutive samples
        if (b < nSamples) {
            float xv = x[b];                          // coalesced b32 load

            // left = floor(x / gs): igs == res/2^20 exactly; 1-ulp quotient
            // error only matters within 1 ulp of a cell boundary where the
            // lerp is continuous -> harmless.
            float q    = xv * igs;
            float fl   = floorf(q);
            int   left = (int)fl;
            float gmin = fl * gs;
            float gmax = gmin + gs;                   // replicate ref rounding
            float den  = gmax - gmin;                 // may differ from gs (ulp of gmin)
            float w    = (xv - gmin) * __builtin_amdgcn_rcpf(den);

            unsigned i0 = ((unsigned)left)        & m;
            unsigned i1 = ((unsigned)(left + 1u)) & m;

            v2f e0 = t[i0];                           // global_load_b64 saddr+scale
            v2f e1 = t[i1];

            float om = 1.0f - w;
            v2f c;
            c.x = e0.x * om + e1.x * w;
            c.y = e0.y * om + e1.y * w;

            // Write-once output: non-temporal so the 256MB stream doesn't
            // evict L2-resident tables. Partial 128B lines from the 16
            // levels merge in L2 before a single HBM writeback.
            __builtin_nontemporal_store(c, &orow[(size_t)b * HE_N_LEVELS]);
        }
    }
}

extern "C" void kernel_launch(void* const* d_in, const int* in_sizes, int n_in,
                              void* d_out, int out_size, void* d_ws, size_t ws_size,
                              hipStream_t stream)
{
    (void)d_ws; (void)ws_size; (void)out_size; (void)n_in;

    const float* x   = (const float*)d_in[0];
    v2f*         out = (v2f*)d_out;

    // Host-side per-level constants, replicating the reference's numpy
    // float32 pipeline:
    //   b   = f32(exp((log(f32(65536)) - log(f32(16))) / 15))
    //   res = f32(floor(f32(16) * b ** f32(i)))
    //   grid_size = 1048576 / res  (python double -> f32 via weak-type promo)
    HEParams p;
    p.hashbits = 0u;
    const float bf = expf((logf(65536.0f) - logf(16.0f)) / 15.0f);
    for (int i = 0; i < HE_N_LEVELS; ++i) {
        float  r  = floorf(16.0f * powf(bf, (float)i));
        double rd = (double)r;
        p.gs[i]   = (float)(1048576.0 / rd);      // IMG_SIZE / res
        p.igs[i]  = (float)(rd / 1048576.0);      // exact: res * 2^-20
        if (rd * rd >= (double)(1u << 19)) p.hashbits |= (1u << i);
        p.tab[i]  = (const v2f*)d_in[1 + i];
    }

    const unsigned nSamples = (unsigned)in_sizes[0];   // B = 2,000,000
    const unsigned spb      = HE_BLOCK * HE_SAMPLES_PER_THREAD;  // samples/block
    dim3 grid((nSamples + spb - 1) / spb, HE_N_LEVELS, 1);
    dim3 block(HE_BLOCK, 1, 1);

    hashenc1d_kernel<<<grid, block, 0, stream>>>(x, out, p, nSamples);
}